// LDStack_56822417326794
// MI455X (gfx1250) — compile-verified
//
#include <hip/hip_runtime.h>

typedef __attribute__((ext_vector_type(2))) float v2f;
typedef __attribute__((ext_vector_type(4))) float v4f;
typedef __attribute__((ext_vector_type(8))) float v8f;

constexpr int T_DIM = 2048;
constexpr int B_DIM = 8;
constexpr int D_DIM = 128;
constexpr int N_DIM = 32;
constexpr int K_DIM = 16;
constexpr int M_DIM = 64;
constexpr int KN    = K_DIM * N_DIM;     // 512
constexpr int CHUNK_L = 32;
constexpr int NCHUNK  = T_DIM / CHUNK_L; // 64
constexpr int LOG2_L  = 5;               // 32 = 2^5

// workspace layout (float offsets)
constexpr int WS_LAM_R  = 0;
constexpr int WS_LAM_I  = WS_LAM_R  + KN;
constexpr int WS_BP_R   = WS_LAM_I  + KN;
constexpr int WS_BP_I   = WS_BP_R   + KN;
constexpr int WS_LAML_R = WS_BP_I   + KN;
constexpr int WS_LAML_I = WS_LAML_R + KN;
constexpr int WS_DMT    = WS_LAML_I + KN;                 // D^T (M,K) scaled 1/K
constexpr int WS_CP4    = WS_DMT    + M_DIM * K_DIM;      // (KN/2, M, 4): {r0,r1,-i0,-i1}/K
constexpr int WS_XK     = WS_CP4    + (KN / 2) * M_DIM * 4; // (T, B, K)
constexpr int WS_P_R    = WS_XK     + T_DIM * B_DIM * K_DIM; // (C, B, KN)
constexpr int WS_P_I    = WS_P_R    + NCHUNK * B_DIM * KN;
constexpr int WS_CA_R   = WS_P_I    + NCHUNK * B_DIM * KN;
constexpr int WS_CA_I   = WS_CA_R   + NCHUNK * B_DIM * KN;

// ---------------- setup: lambda, B', lambda^L, D^T ----------------
__global__ __launch_bounds__(512)
void k_setup(const float* __restrict__ lnr_r, const float* __restrict__ lnr_i,
             const float* __restrict__ lnc_a, const float* __restrict__ lnc_b,
             const int* __restrict__ sidx, int n_real,
             const float* __restrict__ dmat, float* __restrict__ ws) {
  __shared__ float ll_r[KN], ll_i[KN], la_r[KN], la_i[KN];
  int v = threadIdx.x;
  float re, im;
  if (v < n_real) { re = lnr_r[v]; im = lnr_i[v]; }
  else {
    int j = v - n_real;
    re = lnc_a[j >> 1];
    float bb = lnc_b[j >> 1];
    im = (j & 1) ? bb : -bb;
  }
  int idx = sidx[v];
  ll_r[idx] = re; ll_i[idx] = im;
  __syncthreads();
  float mag = expf(ll_r[v]);
  float lr = mag * cosf(ll_i[v]);
  float li = mag * sinf(ll_i[v]);
  la_r[v] = lr; la_i[v] = li;
  ws[WS_LAM_R + v] = lr; ws[WS_LAM_I + v] = li;
  __syncthreads();
  // B'[k,n] = exp(-sum_{j!=n} log1p(-lam_n/lam_j))
  int k = v >> 5, n = v & 31;
  float sr = 0.f, si = 0.f;
  for (int j = 0; j < N_DIM; ++j) {
    if (j == n) continue;
    float ljr = la_r[(k << 5) + j], lji = la_i[(k << 5) + j];
    float d = ljr * ljr + lji * lji;
    float wr = -(lr * ljr + li * lji) / d;
    float wi = -(li * ljr - lr * lji) / d;
    float zr = 1.f + wr, zi = wi;
    sr += 0.5f * logf(zr * zr + zi * zi);
    si += atan2f(zi, zr);
  }
  float em = expf(-sr);
  ws[WS_BP_R + v] = em * cosf(-si);
  ws[WS_BP_I + v] = em * sinf(-si);
  // lam^CHUNK_L by LOG2_L squarings
  float zr = lr, zi = li;
  for (int it = 0; it < LOG2_L; ++it) { float t = zr * zr - zi * zi; zi = 2.f * zr * zi; zr = t; }
  ws[WS_LAML_R + v] = zr; ws[WS_LAML_I + v] = zi;
  // transposed, pre-scaled D matrix: DMT[m*K + k] = D[k,m] / K
  for (int e = v; e < M_DIM * K_DIM; e += 512) {
    int m = e >> 4, kc = e & 15;
    ws[WS_DMT + e] = dmat[kc * M_DIM + m] * (1.0f / K_DIM);
  }
}

// ---------------- C' precompute, pair-interleaved for b128 B-fragments ----------------
__global__ __launch_bounds__(256)
void k_cp(const float* __restrict__ psi_p, const float* __restrict__ Cpn,
          const float* __restrict__ psi_n, const float* __restrict__ Cnn,
          float* __restrict__ ws) {
  int idx = blockIdx.x * 256 + threadIdx.x;  // KN*M = 32768
  int m = idx & (M_DIM - 1);
  int kn = idx >> 6;
  int k = kn >> 5;
  float lr = ws[WS_LAM_R + kn], li = ws[WS_LAM_I + kn];
  float r = sqrtf(lr * lr + li * li);
  float th = atan2f(li, lr);
  float cr = 0.f, ci = 0.f;
  for (int j = 0; j < N_DIM - 1; ++j) {
    float p = (float)(j - (N_DIM - 1));      // -31 .. -1
    float pp = psi_p[(k * (N_DIM - 1) + j) * M_DIM + m];
    float pn = psi_n[(k * (N_DIM - 1) + j) * M_DIM + m];
    float ang = p * th;
    float ca = cosf(ang), sa = sinf(ang);
    float dmag = powf(pp * r, p) - powf(pn * r, p);
    cr += dmag * ca;
    ci += dmag * sa;
  }
  cr += Cpn[k * M_DIM + m] - Cnn[k * M_DIM + m];
  // layout: quad {r_even, r_odd, -i_even, -i_odd}/K at (pair, m)
  int pair = kn >> 1, q = kn & 1;
  int base = ((pair * M_DIM) + m) << 2;
  ws[WS_CP4 + base + q]     =  cr * (1.0f / K_DIM);
  ws[WS_CP4 + base + 2 + q] = -ci * (1.0f / K_DIM);
}

// ---------------- xk = x @ R ----------------
__global__ __launch_bounds__(256)
void k_xk(const float* __restrict__ x, const float* __restrict__ R,
          float* __restrict__ ws) {
  int idx = blockIdx.x * 256 + threadIdx.x;  // T*B*K
  int k = idx & (K_DIM - 1);
  int tb = idx >> 4;
  const float* xp = x + tb * D_DIM;
  float acc = 0.f;
  for (int d = 0; d < D_DIM; ++d) acc += xp[d] * R[d * K_DIM + k];
  ws[WS_XK + idx] = acc;
}

// ---------------- per-chunk partial scan (from zero state) ----------------
__global__ __launch_bounds__(256)
void k_partial(float* __restrict__ ws) {
  int idx = blockIdx.x * 256 + threadIdx.x;  // NCHUNK*B*KN
  int kn = idx & (KN - 1);
  int b  = (idx >> 9) & (B_DIM - 1);
  int c  = idx >> 12;                        // B*KN == 4096 == 2^12
  int k = kn >> 5;
  float lr = ws[WS_LAM_R + kn], li = ws[WS_LAM_I + kn];
  float br = ws[WS_BP_R + kn],  bi = ws[WS_BP_I + kn];
  const float* xk = ws + WS_XK;
  float ar = 0.f, ai = 0.f;
  for (int t = c * CHUNK_L; t < (c + 1) * CHUNK_L; ++t) {
    float xv = xk[(t * B_DIM + b) * K_DIM + k];
    float nr = lr * ar - li * ai + xv * br;
    float ni = lr * ai + li * ar + xv * bi;
    ar = nr; ai = ni;
  }
  ws[WS_P_R + idx] = ar;
  ws[WS_P_I + idx] = ai;
}

// ---------------- chunk carry prefix: Carry[c] = a_{cL-1} ----------------
__global__ __launch_bounds__(256)
void k_prefix(float* __restrict__ ws) {
  int idx = blockIdx.x * 256 + threadIdx.x;  // B*KN = 4096
  int kn = idx & (KN - 1);
  float Lr = ws[WS_LAML_R + kn], Li = ws[WS_LAML_I + kn];
  float cr = 0.f, ci = 0.f;
  for (int c = 0; c < NCHUNK; ++c) {
    int off = c * (B_DIM * KN) + idx;
    ws[WS_CA_R + off] = cr;
    ws[WS_CA_I + off] = ci;
    float pr = ws[WS_P_R + off], pi = ws[WS_P_I + off];
    float nr = Lr * cr - Li * ci + pr;
    float ni = Lr * ci + Li * cr + pi;
    cr = nr; ci = ni;
  }
}

// ---------------- fused scan replay + dual-chain WMMA einsum + Dx + Do ----------------
__global__ __launch_bounds__(128)
void k_main(const float* __restrict__ Do, float* __restrict__ out,
            float* __restrict__ ws) {
  __shared__ float s_tile[2][16][KN];   // 64 KB: s_real / s_imag rows
  int c = blockIdx.x % NCHUNK;
  int b = blockIdx.x / NCHUNK;
  int tid = threadIdx.x;
  int lane = tid & 31;
  int wv = tid >> 5;          // 4 waves; each owns one 16-wide m-tile
  int m0 = wv << 4;
  // recurrence ownership: 4 channels with the same k per thread
  int k  = tid >> 3;
  int nb = (tid & 7) << 2;
  int kn0 = (k << 5) + nb;
  float lr[4], li[4], bpr[4], bpi[4], ar[4], ai[4];
  for (int j = 0; j < 4; ++j) {
    lr[j]  = ws[WS_LAM_R + kn0 + j];  li[j]  = ws[WS_LAM_I + kn0 + j];
    bpr[j] = ws[WS_BP_R  + kn0 + j];  bpi[j] = ws[WS_BP_I  + kn0 + j];
    int off = c * (B_DIM * KN) + b * KN + kn0 + j;
    ar[j] = ws[WS_CA_R + off];        ai[j] = ws[WS_CA_I + off];
  }
  const float* xk  = ws + WS_XK;
  const float* cp4 = ws + WS_CP4;
  const float* dmt = ws + WS_DMT;
  float dov = Do[0];
  int row = lane & 15;
  int khalf = lane >> 4;

  for (int g = 0; g < CHUNK_L / 16; ++g) {
    int t0 = c * CHUNK_L + (g << 4);
    // replay recurrence for 16 timesteps; s_t = a_{t-1}
    for (int tt = 0; tt < 16; ++tt) {
      int t = t0 + tt;
      float xv = xk[(t * B_DIM + b) * K_DIM + k];
      for (int j = 0; j < 4; ++j) {
        s_tile[0][tt][kn0 + j] = ar[j];
        s_tile[1][tt][kn0 + j] = ai[j];
        float nr = lr[j] * ar[j] - li[j] * ai[j] + xv * bpr[j];
        float ni = lr[j] * ai[j] + li[j] * ar[j] + xv * bpi[j];
        ar[j] = nr; ai[j] = ni;
      }
    }
    __syncthreads();
    // y(16xM tile) = Sr @ (Cp_r/K) + Si @ (-Cp_i/K); two independent WMMA chains
    v8f accr = {}, acci = {};
    for (int k0 = 0; k0 < KN; k0 += 4) {
      int kk = k0 + (khalf << 1);
      int pair = kk >> 1;
      v2f afr, afi;
      afr[0] = s_tile[0][row][kk]; afr[1] = s_tile[0][row][kk + 1];
      afi[0] = s_tile[1][row][kk]; afi[1] = s_tile[1][row][kk + 1];
      v4f bv = *(const v4f*)(cp4 + (((pair * M_DIM) + m0 + row) << 2));
      v2f bfr, bfi;
      bfr[0] = bv[0]; bfr[1] = bv[1];
      bfi[0] = bv[2]; bfi[1] = bv[3];
      accr = __builtin_amdgcn_wmma_f32_16x16x4_f32(false, afr, false, bfr,
                                                   (short)0, accr, false, false);
      acci = __builtin_amdgcn_wmma_f32_16x16x4_f32(false, afi, false, bfi,
                                                   (short)0, acci, false, false);
    }
    v8f acc = accr + acci;
    // add mean_k(xk*D) + Do, write real output
    for (int v = 0; v < 8; ++v) {
      int trow = v + (khalf << 3);
      int t = t0 + trow;
      int col = m0 + row;
      const v4f* xr = (const v4f*)(xk + (t * B_DIM + b) * K_DIM);
      const v4f* dr = (const v4f*)(dmt + col * K_DIM);
      v4f dd = xr[0] * dr[0] + xr[1] * dr[1] + xr[2] * dr[2] + xr[3] * dr[3];
      float dx = dd[0] + dd[1] + dd[2] + dd[3];
      out[(t * B_DIM + b) * M_DIM + col] = acc[v] + dx + dov;
    }
    __syncthreads();
  }
}

extern "C" void kernel_launch(void* const* d_in, const int* in_sizes, int n_in,
                              void* d_out, int out_size, void* d_ws, size_t ws_size,
                              hipStream_t stream) {
  (void)n_in; (void)out_size; (void)ws_size;
  const float* x     = (const float*)d_in[0];
  const float* R     = (const float*)d_in[1];
  const float* lnr_r = (const float*)d_in[2];
  const float* lnr_i = (const float*)d_in[3];
  const float* lnc_a = (const float*)d_in[4];
  const float* lnc_b = (const float*)d_in[5];
  const float* psi_p = (const float*)d_in[6];
  const float* Cpn   = (const float*)d_in[7];
  const float* psi_n = (const float*)d_in[8];
  const float* Cnn   = (const float*)d_in[9];
  const float* dmat  = (const float*)d_in[10];
  const float* Do    = (const float*)d_in[11];
  const int*   sidx  = (const int*)d_in[12];
  float* out = (float*)d_out;
  float* ws  = (float*)d_ws;
  int n_real = in_sizes[2];

  k_setup  <<<1, 512, 0, stream>>>(lnr_r, lnr_i, lnc_a, lnc_b, sidx, n_real, dmat, ws);
  k_cp     <<<(KN * M_DIM) / 256, 256, 0, stream>>>(psi_p, Cpn, psi_n, Cnn, ws);
  k_xk     <<<(T_DIM * B_DIM * K_DIM) / 256, 256, 0, stream>>>(x, R, ws);
  k_partial<<<(NCHUNK * B_DIM * KN) / 256, 256, 0, stream>>>(ws);
  k_prefix <<<(B_DIM * KN) / 256, 256, 0, stream>>>(ws);
  k_main   <<<NCHUNK * B_DIM, 128, 0, stream>>>(Do, out, ws);
}